// LocalConnectivity_3796751089942
// MI455X (gfx1250) — compile-verified
//
#include <hip/hip_runtime.h>

#define HW      4096
#define MASKXY  4095
#define TILE    64
#define HALO    5
#define LW      (TILE + 2 * HALO)   // 74
#define LPITCH  (LW + 2)            // 76 floats -> row stride 304B
#define NTHREADS 256

typedef __attribute__((address_space(1))) int g_i32;
typedef __attribute__((address_space(3))) int l_i32;

// out(i,j) = sum_{1<=|dx|+|dy|<=5} w_{|dx|+|dy|} * s((i+dx)&4095, (j+dy)&4095)
// (kernel is symmetric, so roll sign is irrelevant)
__global__ __launch_bounds__(NTHREADS)
void diamond_stencil_f32(const float* __restrict__ in,
                         const float* __restrict__ dw,
                         float* __restrict__ out) {
  __shared__ float tile[LW * LPITCH];

  const int tid   = threadIdx.x;
  const int tileX = blockIdx.x * TILE;
  const int tileY = blockIdx.y * TILE;

  // ---- cooperative halo-tile load with periodic wrap, async global->LDS ----
  for (int idx = tid; idx < LW * LW; idx += NTHREADS) {
    const int ly = idx / LW;
    const int lx = idx - ly * LW;
    const int gy = (tileY + ly - HALO) & MASKXY;
    const int gx = (tileX + lx - HALO) & MASKXY;
    __builtin_amdgcn_global_load_async_to_lds_b32(
        (g_i32*)(in + ((gy << 12) + gx)),
        (l_i32*)(&tile[ly * LPITCH + lx]),
        0, 0);
  }
  __builtin_amdgcn_s_wait_asynccnt(0);
  __syncthreads();

  const float w1 = dw[0], w2 = dw[1], w3 = dw[2], w4 = dw[3], w5 = dw[4];

  const int lc   = tid & (TILE - 1);  // column within tile
  const int rg   = tid >> 6;          // 4 row-groups of 16 rows each
  const int base = rg * 16;

  float acc[16];
#pragma unroll
  for (int t = 0; t < 16; ++t) acc[t] = 0.0f;

  // For each input row, build the 6 |dx|-band horizontal convolutions and
  // scatter them into the 16 per-thread accumulators (fully unrolled so the
  // range checks fold away and acc[] stays in registers).
#pragma unroll
  for (int w = 0; w < 26; ++w) {
    const float* row = &tile[(base + w) * LPITCH + lc];
    const float s0 = row[0], s1 = row[1], s2 = row[2], s3 = row[3], s4 = row[4];
    const float c  = row[5];
    const float s6 = row[6], s7 = row[7], s8 = row[8], s9 = row[9], s10 = row[10];
    const float p1 = s4 + s6;
    const float p2 = s3 + s7;
    const float p3 = s2 + s8;
    const float p4 = s1 + s9;
    const float p5 = s0 + s10;
    float h[6];
    h[0] = w1 * p1 + w2 * p2 + w3 * p3 + w4 * p4 + w5 * p5;  // center weight 0
    h[1] = w1 * c  + w2 * p1 + w3 * p2 + w4 * p3 + w5 * p4;
    h[2] = w2 * c  + w3 * p1 + w4 * p2 + w5 * p3;
    h[3] = w3 * c  + w4 * p1 + w5 * p2;
    h[4] = w4 * c  + w5 * p1;
    h[5] = w5 * c;
    const int r = w - 5;  // input row relative to this thread's first output row
#pragma unroll
    for (int a = 0; a <= 5; ++a) {
      const int t1 = r - a;            // contributes with vertical offset +a
      if (0 <= t1 && t1 < 16) acc[t1] += h[a];
      if (a > 0) {
        const int t2 = r + a;          // contributes with vertical offset -a
        if (0 <= t2 && t2 < 16) acc[t2] += h[a];
      }
    }
  }

  const int gx = tileX + lc;
#pragma unroll
  for (int t = 0; t < 16; ++t) {
    out[((tileY + base + t) << 12) + gx] = acc[t];
  }
}

extern "C" void kernel_launch(void* const* d_in, const int* in_sizes, int n_in,
                              void* d_out, int out_size, void* d_ws, size_t ws_size,
                              hipStream_t stream) {
  const float* grid_spikes = (const float*)d_in[0];  // [4096, 4096] fp32
  const float* dweights    = (const float*)d_in[1];  // [5] fp32
  float* out = (float*)d_out;                        // [4096, 4096] fp32

  dim3 grid(HW / TILE, HW / TILE);  // 64 x 64 tiles
  diamond_stencil_f32<<<grid, dim3(NTHREADS), 0, stream>>>(grid_spikes, dweights, out);
}